// CPN_4492535791617
// MI455X (gfx1250) — compile-verified
//
#include <hip/hip_runtime.h>

// Native clang vectors: required for __builtin_nontemporal_load/store
// (HIP's float2/float4 are structs and are rejected by the builtin).
typedef float v4f __attribute__((ext_vector_type(4)));
typedef float v2f __attribute__((ext_vector_type(2)));

// Problem constants (fixed in the reference module).
#define HH      512
#define WW      512
#define NB      6          // num_buckets
#define NCH     (2 * NB)   // 12 channels
#define NLOOPS  4
#define NDET    4096
#define NSAMP   128
#define NBATCH  4
#define NPAIRS  (NDET * NSAMP)   // 524,288 independent gather chains

// ---------------------------------------------------------------------------
// Pass 1: transpose refinement (B, 12, H, W) -> (B, H, W, 12).
//
// Each thread handles 4 consecutive x-pixels of one (b, y) row:
//   - 12 x global_load_b128 nontemporal (source is dead after this pass;
//     keep it from evicting the transposed table out of L2)
//   - 12 x global_store_b128 regular temporal (this IS the hot table)
// W=H=512 are powers of two -> all index math is shifts/masks, no div/mod.
// ~100 MB streamed at 23.3 TB/s ~= 5 us; it collapses each random gather
// from 4 separate 1MB-apart planes into one 48-byte AoS pixel record.
// ---------------------------------------------------------------------------
#define QUADS_PER_ROW (WW / 4)                       // 128
#define NT_THREADS    (NBATCH * HH * QUADS_PER_ROW)  // 262,144

__global__ void CPN_transpose_ref(const float* __restrict__ src,
                                  float* __restrict__ dst) {
    const int t = blockIdx.x * blockDim.x + threadIdx.x;
    if (t >= NT_THREADS) return;
    const int xq = (t & (QUADS_PER_ROW - 1)) << 2;   // x0 = quad * 4
    const int y  = (t >> 7) & (HH - 1);
    const int b  = t >> 16;                          // 7 + 9 bits below

    v4f v[NCH];
#pragma unroll
    for (int c = 0; c < NCH; ++c) {
        const v4f* p = (const v4f*)(src +
            (((size_t)b * NCH + c) * HH + y) * WW + xq);
        v[c] = __builtin_nontemporal_load(p);        // th: NT (streamed once)
    }

    // Four 48-byte records, each written as 3x b128 (16B aligned).
#pragma unroll
    for (int i = 0; i < 4; ++i) {
        v4f* out = (v4f*)(dst +
            (((size_t)b * HH + y) * WW + (xq + i)) * NCH);
        out[0] = (v4f){v[0][i], v[1][i],  v[2][i],  v[3][i]};
        out[1] = (v4f){v[4][i], v[5][i],  v[6][i],  v[7][i]};
        out[2] = (v4f){v[8][i], v[9][i], v[10][i], v[11][i]};
    }
}

// ---------------------------------------------------------------------------
// Pass 2: the dependent refinement chain. One lane per (det, sample) pair.
//
// Per-sample bucket math (identical to reference):
//   base   = sampling[s] * NB,  base_i = floor(base), frac = base - base_i
//   off=-1: d = |base_i-1 - base| = 1 + frac >= 1  =>  w == 0 exactly
//           (w = where(d>1, 0, 1-d); at frac==0, 1-d == 0).  Its term is
//           0 * finite == 0 and (0 + x) == x bit-exactly -> gather skipped.
//   off= 0: w0 = 1 - frac   (bucket bi0 = base_i % NB)
//   off=+1: w1 = frac       (bucket bi1 = (base_i+1) % NB)
// base_i is provably in [0, NB] so "% NB" == compare-subtract (no idiv).
//
// Each loop: rx = clamp(rintf(dx)) (rintf == RNE == jnp.round), gather the
// two (x,y) channel pairs as aligned b64 loads (usually one cache line in
// the 48B AoS record), dx = rx + w0*c0x + w1*c1x.  16K wave32 waves hide
// the 4 dependent L2 round trips.  det/out are nontemporal streams.
// ---------------------------------------------------------------------------
template <bool TRANSPOSED>
__global__ void CPN_refine(const float* __restrict__ det_in,
                           const float* __restrict__ ref,   // transposed or native
                           const float* __restrict__ sampling,
                           const int*   __restrict__ b_idx,
                           float* __restrict__ out) {
    const int pair = blockIdx.x * blockDim.x + threadIdx.x;
    if (pair >= NPAIRS) return;
    const int s = pair & (NSAMP - 1);      // sample id
    const int d = pair >> 7;               // detection id (NSAMP == 128)
    const int bb = b_idx[d];               // uniform across each 128-lane group

    // Per-sample weights / bucket indices (cheap VALU; no LDS needed).
    const float base  = sampling[s] * (float)NB;       // in [0, 6)
    const float bif   = floorf(base);
    int basei = (int)bif;                              // in [0, 6] worst case
    if (basei >= NB) basei -= NB;                      // == basei % NB
    const float d0 = fabsf(bif - base);
    const float w0 = (d0 > 1.0f) ? 0.0f : 1.0f - d0;
    const float d1 = fabsf((bif + 1.0f) - base);
    const float w1 = (d1 > 1.0f) ? 0.0f : 1.0f - d1;
    const int bi0 = basei;
    int bi1 = basei + 1;
    if (bi1 >= NB) bi1 = 0;                            // == (basei+1) % NB

    const v2f dv = __builtin_nontemporal_load(((const v2f*)det_in) + pair);
    float dx = dv.x, dy = dv.y;

#pragma unroll
    for (int l = 0; l < NLOOPS; ++l) {
        // round (RNE, matches jnp.round), clip, truncate-to-int
        const float rx = fminf(fmaxf(rintf(dx), 0.0f), (float)(WW - 1));
        const float ry = fminf(fmaxf(rintf(dy), 0.0f), (float)(HH - 1));
        const int xi = (int)rx;
        const int yi = (int)ry;

        float c0x, c0y, c1x, c1y;
        if (TRANSPOSED) {
            // 48B AoS pixel record; both b64 loads usually hit one cache line.
            const float* p = ref + ((size_t)((bb << 9) + yi) * WW + xi) * NCH;
            const v2f q0 = *(const v2f*)(p + 2 * bi0);
            const v2f q1 = *(const v2f*)(p + 2 * bi1);
            c0x = q0.x; c0y = q0.y; c1x = q1.x; c1y = q1.y;
        } else {
            // Native (B, 12, H, W): four scalar loads from four planes.
            const size_t plane = (size_t)HH * WW;
            const float* p = ref + (size_t)bb * NCH * plane + (size_t)yi * WW + xi;
            c0x = p[(size_t)(2 * bi0)     * plane];
            c0y = p[(size_t)(2 * bi0 + 1) * plane];
            c1x = p[(size_t)(2 * bi1)     * plane];
            c1y = p[(size_t)(2 * bi1 + 1) * plane];
        }

        dx = rx + (c0x * w0 + c1x * w1);
        dy = ry + (c0y * w0 + c1y * w1);
    }

    const v2f res = {dx, dy};
    __builtin_nontemporal_store(res, ((v2f*)out) + pair);
}

// ---------------------------------------------------------------------------
// Inputs (setup_inputs order):
//   d_in[0] det_indices f32 (4096,128,2)   d_in[1] refinement f32 (4,12,512,512)
//   d_in[2] sampling    f32 (128,)         d_in[3] b          i32 (4096,)
//   d_in[4..7] scalar ints (num_loops, num_buckets, height, width) — fixed,
//   baked in as compile-time constants above.
// Output: f32 (4096,128,2).
// ---------------------------------------------------------------------------
extern "C" void kernel_launch(void* const* d_in, const int* in_sizes, int n_in,
                              void* d_out, int out_size, void* d_ws, size_t ws_size,
                              hipStream_t stream) {
    const float* det  = (const float*)d_in[0];
    const float* ref  = (const float*)d_in[1];
    const float* samp = (const float*)d_in[2];
    const int*   b    = (const int*)d_in[3];
    float*       out  = (float*)d_out;

    const size_t need = (size_t)NBATCH * HH * WW * NCH * sizeof(float); // 48 MB

    const int threads = 256;                       // 8 wave32 waves / block
    const int gather_blocks = (NPAIRS + threads - 1) / threads;

    if (d_ws != nullptr && ws_size >= need) {
        float* ref_t = (float*)d_ws;
        CPN_transpose_ref<<<(NT_THREADS + threads - 1) / threads, threads, 0, stream>>>(ref, ref_t);
        CPN_refine<true><<<gather_blocks, threads, 0, stream>>>(det, ref_t, samp, b, out);
    } else {
        CPN_refine<false><<<gather_blocks, threads, 0, stream>>>(det, ref, samp, b, out);
    }
}